// BackwardConv2D_33457795236303
// MI455X (gfx1250) — compile-verified
//
#include <hip/hip_runtime.h>

typedef float v2f __attribute__((ext_vector_type(2)));
typedef float v8f __attribute__((ext_vector_type(8)));

#define Hh   32
#define Ww   32
#define CIN  32
#define COUT 64
#define NOUT 32

// ---------------------------------------------------------------------------
// Main kernel: one wave per (pass, b, y, x). Computes OUT[ci=32, n=32] via
// 4 accumulators of V_WMMA_F32_16X16X4_F32, contracting over (kh,kw,co)=576.
// ---------------------------------------------------------------------------
__global__ __launch_bounds__(128) void conv_adjoint_wmma(
    const float* __restrict__ w_u, const float* __restrict__ w_l,
    const float* __restrict__ ker,
    float* __restrict__ out_wu, float* __restrict__ out_wl)
{
    const unsigned tid  = threadIdx.x;
    const unsigned lane = tid & 31u;
    const unsigned gw   = blockIdx.x * 4u + (tid >> 5);   // 0 .. 16383
    const unsigned pass = gw >> 13;                        // 0=upper, 1=lower
    const unsigned rem  = gw & 8191u;
    const unsigned b    = rem >> 10;                       // batch 0..7
    const unsigned p    = rem & 1023u;                     // pixel 0..1023
    const int y = (int)(p >> 5);
    const int x = (int)(p & 31u);

    const float* __restrict__ Wi = pass ? w_l : w_u;
    float* __restrict__       Wo = pass ? out_wl : out_wu;

    const unsigned m  = lane & 15u;          // row (A: ci, B: n) within half
    const unsigned k2 = (lane >> 4) * 2u;    // K pair offset: 0 or 2

    v8f acc00 = {}; v8f acc01 = {}; v8f acc10 = {}; v8f acc11 = {};

    const float* inB = Wi + (size_t)b * (size_t)(Hh * Ww * COUT * NOUT);

#pragma unroll
    for (int tap = 0; tap < 9; ++tap) {
        const int kh = tap / 3, kw = tap % 3;
        const int yy = y + 1 - kh;
        const int xx = x + 1 - kw;
        if ((unsigned)yy >= 32u || (unsigned)xx >= 32u) continue;  // OOB -> zero

        const float* __restrict__ in_row = inB + (size_t)((yy * 32 + xx) * COUT) * NOUT;
        const float* __restrict__ krow   = ker + (size_t)((kh * 3 + kw) * CIN) * COUT;

#pragma unroll 4
        for (int cb = 0; cb < 16; ++cb) {
            const int co = cb * 4 + (int)k2;            // this lane's K columns
            // A fragments (16x4, fp32): A[m][k] = K[kh,kw, ci_base+m, co_base+k]
            v2f a0 = *(const v2f*)(krow + (m)        * COUT + co);  // ci 0..15
            v2f a1 = *(const v2f*)(krow + (m + 16u)  * COUT + co);  // ci 16..31
            // B fragments (4x16, fp32): B[k][n] = IN[(yy,xx, co_base+k), n_base+n]
            const float* bp = in_row + (size_t)co * NOUT + m;
            v2f b0, b1;
            b0.x = bp[0];          b0.y = bp[NOUT];          // n_base = 0
            b1.x = bp[16];         b1.y = bp[NOUT + 16];     // n_base = 16
            acc00 = __builtin_amdgcn_wmma_f32_16x16x4_f32(false, a0, false, b0, (short)0, acc00, false, false);
            acc01 = __builtin_amdgcn_wmma_f32_16x16x4_f32(false, a0, false, b1, (short)0, acc01, false, false);
            acc10 = __builtin_amdgcn_wmma_f32_16x16x4_f32(false, a1, false, b0, (short)0, acc10, false, false);
            acc11 = __builtin_amdgcn_wmma_f32_16x16x4_f32(false, a1, false, b1, (short)0, acc11, false, false);
        }
    }

    // Store: C/D 16x16 layout -> VGPR r, lane L holds (ci = r + 8*(L>=16), n = L&15)
    float* __restrict__ outp =
        Wo + (((size_t)b * 1024u + (unsigned)(y * 32 + x)) * CIN) * NOUT;
    const unsigned mo = (lane >> 4) * 8u;
#pragma unroll
    for (int r = 0; r < 8; ++r) {
        const unsigned ci0 = (unsigned)r + mo;
        outp[(ci0)       * NOUT + m]        = acc00[r];
        outp[(ci0)       * NOUT + 16u + m]  = acc01[r];
        outp[(ci0 + 16u) * NOUT + m]        = acc10[r];
        outp[(ci0 + 16u) * NOUT + 16u + m]  = acc11[r];
    }
}

// ---------------------------------------------------------------------------
// Bias path: b_new[b,n] = sum_{h,w,c} w[b,(h,w,c),n]*bias[c] + b[b,n]
// ---------------------------------------------------------------------------
__global__ void bias_init(const float* __restrict__ b_u, const float* __restrict__ b_l,
                          float* __restrict__ out_bu, float* __restrict__ out_bl)
{
    const unsigned i = threadIdx.x;     // 256 threads
    out_bu[i] = b_u[i];
    out_bl[i] = b_l[i];
}

__global__ __launch_bounds__(256) void bias_reduce(
    const float* __restrict__ w_u, const float* __restrict__ w_l,
    const float* __restrict__ bias,
    float* __restrict__ out_bu, float* __restrict__ out_bl)
{
    // grid = 2 * 8 * 64 blocks; each block reduces a 1024-wide flat slice
    const unsigned blk  = blockIdx.x;
    const unsigned pass = blk >> 9;
    const unsigned rem  = blk & 511u;
    const unsigned b    = rem >> 6;
    const unsigned s    = rem & 63u;

    const float* __restrict__ Wi = pass ? w_l : w_u;
    float* __restrict__       Bo = pass ? out_bl : out_bu;

    const unsigned t = threadIdx.x;
    const unsigned n = t & 31u;
    const unsigned g = t >> 5;          // 0..7
    const unsigned f0 = s * 1024u;
    const float* base = Wi + (size_t)b * 65536u * (size_t)NOUT;

    float acc = 0.f;
    for (unsigned f = f0 + g; f < f0 + 1024u; f += 8u)
        acc += base[(size_t)f * NOUT + n] * bias[f & 63u];

    __shared__ float sm[256];
    sm[t] = acc;
    __syncthreads();
    if (g == 0) {
        float tot = sm[n];
#pragma unroll
        for (int j = 1; j < 8; ++j) tot += sm[j * 32 + n];
        atomicAdd(&Bo[b * 32u + n], tot);
    }
}

// ---------------------------------------------------------------------------
extern "C" void kernel_launch(void* const* d_in, const int* in_sizes, int n_in,
                              void* d_out, int out_size, void* d_ws, size_t ws_size,
                              hipStream_t stream)
{
    const float* w_u  = (const float*)d_in[0];
    const float* b_u  = (const float*)d_in[1];
    const float* w_l  = (const float*)d_in[2];
    const float* b_l  = (const float*)d_in[3];
    const float* ker  = (const float*)d_in[4];
    const float* bias = (const float*)d_in[5];

    float* out = (float*)d_out;
    const size_t NW = (size_t)8 * 32768 * 32;   // elements in each w output
    float* out_wu = out;
    float* out_bu = out + NW;
    float* out_wl = out + NW + 256;
    float* out_bl = out + 2 * NW + 256;

    // bias path (init then atomic partial sums; serialized on the same stream)
    bias_init<<<1, 256, 0, stream>>>(b_u, b_l, out_bu, out_bl);
    bias_reduce<<<1024, 256, 0, stream>>>(w_u, w_l, bias, out_bu, out_bl);

    // main adjoint-conv GEMM: 16384 wave-jobs, 4 waves / block
    conv_adjoint_wmma<<<4096, 128, 0, stream>>>(w_u, w_l, ker, out_wu, out_wl);
}